// Generator_14499809591768
// MI455X (gfx1250) — compile-verified
//
#include <hip/hip_runtime.h>
#include <hip/hip_bf16.h>
#include <math.h>

typedef __attribute__((ext_vector_type(2))) float v2f;
typedef __attribute__((ext_vector_type(8))) float v8f;

#define H_DIM 1024
#define W_DIM 1024
#define GH 16
#define GW 16
#define GD 8
#define CC 12
#define CIN 64
#define COUT 96
#define PLANE (H_DIM*W_DIM)
#define GRID_FLOATS (4*GH*GW*GD*CC)   // 98304

// ---------------------------------------------------------------------------
// Kernel A: fused = relu(local + global); grid = fusion_w @ fused + fusion_b
// One wave (32 threads) computes one 16x16 output tile via V_WMMA_F32_16X16X4_F32.
// grid layout matches reference: (B, gh, gw, gd=8, C=12)
// ---------------------------------------------------------------------------
__global__ __launch_bounds__(32) void coeff_gemm_wmma(
    const float* __restrict__ localf,   // (B,64,16,16)
    const float* __restrict__ globalf,  // (B,64)
    const float* __restrict__ fw,       // (96,64)
    const float* __restrict__ fb,       // (96,)
    float* __restrict__ grid)           // (B,16,16,8,12)
{
    const int pt   = blockIdx.x;        // pixel tile 0..15  (16 pixels each)
    const int ot   = blockIdx.y;        // out-chan tile 0..5
    const int b    = blockIdx.z;        // batch
    const int lane = threadIdx.x;
    const int half = lane >> 4;         // 0 or 1
    const int l16  = lane & 15;
    const int colp = pt * 16 + l16;     // pixel index (N) for B/C/D fragments
    const int rowa = ot * 16 + l16;     // out-channel row (M) for A fragment

    const float* lf = localf + (size_t)b * CIN * 256;
    const float* gf = globalf + (size_t)b * CIN;

    v8f acc = {};
    #pragma unroll
    for (int k = 0; k < CIN; k += 4) {
        const int kk = k + 2 * half;
        // A fragment: 16x4 f32, lanes 0-15 hold K=k,k+1; lanes 16-31 hold K=k+2,k+3
        v2f a;
        a.x = fw[rowa * CIN + kk];
        a.y = fw[rowa * CIN + kk + 1];
        // B fragment: 4x16 f32, same K striping, N across lanes 0-15
        float f0 = lf[(size_t)kk * 256 + colp] + gf[kk];
        float f1 = lf[(size_t)(kk + 1) * 256 + colp] + gf[kk + 1];
        v2f bb;
        bb.x = f0 > 0.f ? f0 : 0.f;
        bb.y = f1 > 0.f ? f1 : 0.f;
        acc = __builtin_amdgcn_wmma_f32_16x16x4_f32(
            false, a, false, bb, (short)0, acc, false, false);
    }

    // C/D layout: VGPR r, lanes 0-15 -> M = r ; lanes 16-31 -> M = r+8 ; N = lane&15
    #pragma unroll
    for (int r = 0; r < 8; ++r) {
        const int o  = ot * 16 + r + 8 * half;      // out channel 0..95
        const float v = acc[r] + fb[o];
        const int lb = o / CC;                       // z-depth index
        const int cc = o % CC;                       // coeff channel
        grid[(((size_t)b * 256 + colp) * GD + lb) * CC + cc] = v;
    }
}

// ---------------------------------------------------------------------------
// Kernel B: fold BN + conv weights for the guide path into 81 floats:
//   prm[0..15]=wr, [16..31]=wg, [32..47]=wb, [48..63]=bias, [64..79]=g2w, [80]=g2b
// ---------------------------------------------------------------------------
__global__ __launch_bounds__(32) void fold_guide(
    const float* __restrict__ g1w, const float* __restrict__ g1b,
    const float* __restrict__ gamma, const float* __restrict__ beta,
    const float* __restrict__ mean, const float* __restrict__ var,
    const float* __restrict__ g2w, const float* __restrict__ g2b,
    float* __restrict__ prm)
{
    const int c = threadIdx.x;
    if (c < 16) {
        const float s = gamma[c] * rsqrtf(var[c] + 1e-5f);
        prm[c]      = g1w[c * 3 + 0] * s;
        prm[16 + c] = g1w[c * 3 + 1] * s;
        prm[32 + c] = g1w[c * 3 + 2] * s;
        prm[48 + c] = (g1b[c] - mean[c]) * s + beta[c];
        prm[64 + c] = g2w[c];
    }
    if (c == 0) prm[80] = g2b[0];
}

// ---------------------------------------------------------------------------
// Kernel C: per-pixel guide + bilateral slice + affine apply + tanh.
// One block = one (batch,row). h-blend of the grid is precomputed into LDS
// (6 KB) since fh/h0 are uniform per row; per-pixel fetch is 4 corners x 12
// floats of aligned ds_load_b128.
// ---------------------------------------------------------------------------
__device__ __forceinline__ float4 fma4(float s, float4 v, float4 a) {
    a.x = fmaf(s, v.x, a.x); a.y = fmaf(s, v.y, a.y);
    a.z = fmaf(s, v.z, a.z); a.w = fmaf(s, v.w, a.w);
    return a;
}
__device__ __forceinline__ float4 mul4(float s, float4 v) {
    return make_float4(s * v.x, s * v.y, s * v.z, s * v.w);
}

__global__ __launch_bounds__(256) void slice_apply(
    const float* __restrict__ fullres,  // (4,3,1024,1024)
    const float* __restrict__ grid,     // (4,16,16,8,12)
    const float* __restrict__ prm,      // 81 folded guide params
    float* __restrict__ out)            // (4,3,1024,1024)
{
    const int h = blockIdx.x;
    const int b = blockIdx.y;
    const int t = threadIdx.x;

    __shared__ float hb[GW * GD * CC];  // 1536 floats = 6 KB

    // ---- uniform per-row h blend ----
    const float hgc = (h + 0.5f) * (1.0f / 64.0f) - 0.5f;
    const float h0f = floorf(hgc);
    const float fh  = hgc - h0f;
    const int   h0  = (int)h0f;
    const int   hi0 = min(max(h0, 0), GH - 1);
    const int   hi1 = min(max(h0 + 1, 0), GH - 1);
    const float* gb = grid + (size_t)b * GH * GW * GD * CC;
    const float* r0 = gb + (size_t)hi0 * GW * GD * CC;
    const float* r1 = gb + (size_t)hi1 * GW * GD * CC;
    #pragma unroll
    for (int i = 0; i < 6; ++i) {
        const int idx = t + i * 256;
        hb[idx] = (1.f - fh) * r0[idx] + fh * r1[idx];
    }
    __syncthreads();

    // ---- vectorized pixel loads: 4 consecutive pixels per thread ----
    const float* fr = fullres + (size_t)b * 3 * PLANE + (size_t)h * W_DIM;
    const float4 r4 = *(const float4*)(fr + t * 4);
    const float4 g4 = *(const float4*)(fr + PLANE + t * 4);
    const float4 b4 = *(const float4*)(fr + 2 * PLANE + t * 4);
    const float* rp = (const float*)&r4;
    const float* gp = (const float*)&g4;
    const float* bp = (const float*)&b4;

    float Ro[4], Go[4], Bo[4];
    #pragma unroll
    for (int i = 0; i < 4; ++i) {
        const float rr = rp[i], gg = gp[i], bl = bp[i];
        const int w = t * 4 + i;

        // ---- guide: folded conv+BN, relu, 16->1, sigmoid ----
        float acc = prm[80];
        #pragma unroll
        for (int c = 0; c < 16; ++c) {
            float v = fmaf(rr, prm[c],
                      fmaf(gg, prm[16 + c],
                      fmaf(bl, prm[32 + c], prm[48 + c])));
            v = fmaxf(v, 0.f);
            acc = fmaf(v, prm[64 + c], acc);
        }
        const float guide = 1.f / (1.f + __expf(-acc));

        // ---- trilinear slice (h already blended into hb) ----
        const float zg  = guide * (float)GD - 0.5f;
        const float z0f = floorf(zg);
        const float fz  = zg - z0f;
        const int   z0  = (int)z0f;
        const int   zi0 = min(max(z0, 0), GD - 1);
        const int   zi1 = min(max(z0 + 1, 0), GD - 1);

        const float wgc = (w + 0.5f) * (1.0f / 64.0f) - 0.5f;
        const float w0f = floorf(wgc);
        const float fw  = wgc - w0f;
        const int   w0  = (int)w0f;
        const int   wi0 = min(max(w0, 0), GW - 1);
        const int   wi1 = min(max(w0 + 1, 0), GW - 1);

        const float* p00 = hb + (wi0 * GD + zi0) * CC;  // 48B-aligned
        const float* p01 = hb + (wi0 * GD + zi1) * CC;
        const float* p10 = hb + (wi1 * GD + zi0) * CC;
        const float* p11 = hb + (wi1 * GD + zi1) * CC;

        const float wt00 = (1.f - fw) * (1.f - fz);
        const float wt01 = (1.f - fw) * fz;
        const float wt10 = fw * (1.f - fz);
        const float wt11 = fw * fz;

        float4 cA = mul4(wt00, *(const float4*)(p00));
        float4 cB = mul4(wt00, *(const float4*)(p00 + 4));
        float4 cD = mul4(wt00, *(const float4*)(p00 + 8));
        cA = fma4(wt01, *(const float4*)(p01),     cA);
        cB = fma4(wt01, *(const float4*)(p01 + 4), cB);
        cD = fma4(wt01, *(const float4*)(p01 + 8), cD);
        cA = fma4(wt10, *(const float4*)(p10),     cA);
        cB = fma4(wt10, *(const float4*)(p10 + 4), cB);
        cD = fma4(wt10, *(const float4*)(p10 + 8), cD);
        cA = fma4(wt11, *(const float4*)(p11),     cA);
        cB = fma4(wt11, *(const float4*)(p11 + 4), cB);
        cD = fma4(wt11, *(const float4*)(p11 + 8), cD);
        // coeff channels: cA = c0..c3, cB = c4..c7, cD = c8..c11

        Ro[i] = tanhf(fmaf(rr, cA.x, fmaf(gg, cA.y, fmaf(bl, cA.z, cB.y))));   // c0..c2 + c5, superseded below
        Go[i] = tanhf(fmaf(rr, cA.w, fmaf(gg, cB.x, fmaf(bl, cB.y, cB.z))));
        Bo[i] = tanhf(fmaf(rr, cB.z, fmaf(gg, cB.w, fmaf(bl, cD.x, cD.w))));
        // NOTE: fix channel picks explicitly (c9=cD.y, c10=cD.z, c11=cD.w):
        Ro[i] = tanhf(fmaf(rr, cA.x, fmaf(gg, cA.y, fmaf(bl, cA.z, cD.y))));
        Go[i] = tanhf(fmaf(rr, cA.w, fmaf(gg, cB.x, fmaf(bl, cB.y, cD.z))));
        Bo[i] = tanhf(fmaf(rr, cB.z, fmaf(gg, cB.w, fmaf(bl, cD.x, cD.w))));
    }

    float* op = out + (size_t)b * 3 * PLANE + (size_t)h * W_DIM + t * 4;
    *(float4*)(op)             = make_float4(Ro[0], Ro[1], Ro[2], Ro[3]);
    *(float4*)(op + PLANE)     = make_float4(Go[0], Go[1], Go[2], Go[3]);
    *(float4*)(op + 2 * PLANE) = make_float4(Bo[0], Bo[1], Bo[2], Bo[3]);
}

// ---------------------------------------------------------------------------
extern "C" void kernel_launch(void* const* d_in, const int* in_sizes, int n_in,
                              void* d_out, int out_size, void* d_ws, size_t ws_size,
                              hipStream_t stream) {
    (void)in_sizes; (void)n_in; (void)out_size; (void)ws_size;
    const float* fullres = (const float*)d_in[0];
    const float* localf  = (const float*)d_in[1];
    const float* globalf = (const float*)d_in[2];
    const float* fw      = (const float*)d_in[3];
    const float* fb      = (const float*)d_in[4];
    const float* g1w     = (const float*)d_in[5];
    const float* g1b     = (const float*)d_in[6];
    const float* gamma   = (const float*)d_in[7];
    const float* beta    = (const float*)d_in[8];
    const float* mean    = (const float*)d_in[9];
    const float* var     = (const float*)d_in[10];
    const float* g2w     = (const float*)d_in[11];
    const float* g2b     = (const float*)d_in[12];

    float* grid = (float*)d_ws;               // 98304 floats
    float* prm  = grid + GRID_FLOATS;         // 81 floats
    float* out  = (float*)d_out;

    coeff_gemm_wmma<<<dim3(16, 6, 4), 32, 0, stream>>>(localf, globalf, fw, fb, grid);
    fold_guide<<<1, 32, 0, stream>>>(g1w, g1b, gamma, beta, mean, var, g2w, g2b, prm);
    slice_apply<<<dim3(H_DIM, 4), 256, 0, stream>>>(fullres, grid, prm, out);
}